// ODE_RNN_25348896981674
// MI455X (gfx1250) — compile-verified
//
#include <hip/hip_runtime.h>

// ODE Euler scan, 32768 rows x 2048 steps x 3 components (fp32 out).
// Memory-bound on the 805MB output stream -> stage per-row step chunks in LDS
// and stream them out with CDNA5 async LDS->global stores (ASYNCcnt),
// double-buffered, non-temporal. No barriers needed: each thread owns its row.

#define ODE_STEPS   2048
#define ODE_CH      16                 // steps staged per chunk
#define ODE_NCH     (ODE_STEPS/ODE_CH) // 128 chunks
#define ODE_BLOCK   128                // threads per block (4 waves)
#define ODE_ROWF    52                 // padded floats per row-chunk in LDS (48 used)
#define ODE_ROWB    (ODE_ROWF*4)       // 208 bytes (16B aligned stride)
#define ODE_BUFB    (ODE_BLOCK*ODE_ROWB)
#define ODE_SEGB    (ODE_CH*3*4)       // 192 bytes of real data per row-chunk
#define ODE_NB128   (ODE_SEGB/16)      // 12 async b128 pieces per row-chunk

#define C_KAIC 3.4f
#define C_KAIA 1.3f
#define K_HALF 0.43f
#define H_STEP 0.01f
#define CLIPV  100000.0f

__global__ __launch_bounds__(ODE_BLOCK)
void ode_scan_async_kernel(const float* __restrict__ x,
                           const float* __restrict__ kern,
                           float* __restrict__ out, int n)
{
    __shared__ __align__(16) float smem[2][ODE_BLOCK][ODE_ROWF];

    const int tid = threadIdx.x;
    const int row = blockIdx.x * ODE_BLOCK + tid;
    if (row >= n) return;

    // Uniform coefficients -> scalar loads / SGPRs.
    float K0[8], KA[8];
#pragma unroll
    for (int i = 0; i < 8; ++i) { K0[i] = kern[i]; KA[i] = kern[8 + i]; }

    // Initial state (coalesced: 12B/lane contiguous).
    float T = x[row * 3 + 0];
    float D = x[row * 3 + 1];
    float S = x[row * 3 + 2];

    // LDS byte offset of our staging buffers (flat-ptr low 32 bits == LDS addr).
    const unsigned ldsBase = (unsigned)(unsigned long long)(size_t)(&smem[0][0][0]);
    const unsigned long long outB = (unsigned long long)out;           // SGPR pair
    const unsigned rowByteBase = (unsigned)row * (unsigned)(ODE_STEPS * 3 * 4);

    for (int c = 0; c < ODE_NCH; ++c) {
        const int buf = c & 1;
        float* rp = &smem[buf][tid][0];

#pragma unroll
        for (int s = 0; s < ODE_CH; ++s) {
            // A = max(0, C_KAIA - 2*S); frac = A/(K_HALF+A)
            float A    = fmaxf(0.0f, fmaf(-2.0f, S, C_KAIA));
            float frac = A * __builtin_amdgcn_rcpf(K_HALF + A);

            // rates = K0 + KA*frac  (order: Kut Ktd Ksd Kus Ktu Kdt Kds Ksu)
            float r0 = fmaf(KA[0], frac, K0[0]);  // Kut
            float r1 = fmaf(KA[1], frac, K0[1]);  // Ktd
            float r2 = fmaf(KA[2], frac, K0[2]);  // Ksd
            float r3 = fmaf(KA[3], frac, K0[3]);  // Kus
            float r4 = fmaf(KA[4], frac, K0[4]);  // Ktu
            float r5 = fmaf(KA[5], frac, K0[5]);  // Kdt
            float r6 = fmaf(KA[6], frac, K0[6]);  // Kds
            float r7 = fmaf(KA[7], frac, K0[7]);  // Ksu

            float U = C_KAIC - (T + D + S);

            float y0 = r0 * U;
            y0 = fmaf(r5, D, y0); y0 = fmaf(-r4, T, y0); y0 = fmaf(-r1, T, y0);
            float y1 = r1 * T;
            y1 = fmaf(r2, S, y1); y1 = fmaf(-r5, D, y1); y1 = fmaf(-r6, D, y1);
            float y2 = r3 * U;
            y2 = fmaf(r6, D, y2); y2 = fmaf(-r7, S, y2); y2 = fmaf(-r2, S, y2);

            y0 = fminf(fmaxf(y0, -CLIPV), CLIPV);
            y1 = fminf(fmaxf(y1, -CLIPV), CLIPV);
            y2 = fminf(fmaxf(y2, -CLIPV), CLIPV);

            T = fmaf(H_STEP, y0, T);
            D = fmaf(H_STEP, y1, D);
            S = fmaf(H_STEP, y2, S);

            rp[3 * s + 0] = T;
            rp[3 * s + 1] = D;
            rp[3 * s + 2] = S;
        }

        // Our own LDS writes must land before the async engine reads them.
        asm volatile("s_wait_dscnt 0" ::: "memory");

        const unsigned gb = rowByteBase + (unsigned)c * (unsigned)ODE_SEGB;
        const unsigned lb = ldsBase + (unsigned)buf * (unsigned)ODE_BUFB
                                    + (unsigned)tid * (unsigned)ODE_ROWB;
#pragma unroll
        for (int j = 0; j < ODE_NB128; ++j) {
            asm volatile(
                "global_store_async_from_lds_b128 %0, %1, %2 th:TH_STORE_NT"
                :: "v"(gb + 16u * j), "v"(lb + 16u * j), "s"(outB)
                : "memory");
        }
        // Allow this chunk's 12 stores to stay in flight; previous buffer done.
        asm volatile("s_wait_asynccnt 12" ::: "memory");
    }
    asm volatile("s_wait_asynccnt 0" ::: "memory");
}

extern "C" void kernel_launch(void* const* d_in, const int* in_sizes, int n_in,
                              void* d_out, int out_size, void* d_ws, size_t ws_size,
                              hipStream_t stream) {
    const float* x    = (const float*)d_in[0];
    const float* kern = (const float*)d_in[1];
    float* out        = (float*)d_out;
    const int n = in_sizes[0] / 3;                 // 32768 rows
    const int grid = (n + ODE_BLOCK - 1) / ODE_BLOCK;
    ode_scan_async_kernel<<<grid, ODE_BLOCK, 0, stream>>>(x, kern, out, n);
}